// C51Agent_38955353375480
// MI455X (gfx1250) — compile-verified
//
#include <hip/hip_runtime.h>
#include <stdint.h>

// C51 categorical projection for MI455X (gfx1250).
// Memory-bound (~436 MB @ 23.3 TB/s => ~19us floor). Uses CDNA5 async
// global<->LDS copies (ASYNCcnt) for coalesced b128 data movement and
// ds_add_f32 for the row-local scatter. One thread owns one row.

#define ATOMS   51
#define RPB     128                     // rows per block == threads per block
#define VMINF  -10.0f
#define VMAXF   10.0f
#define DELTAF  0.4f                    // (VMAX - VMIN) / (ATOMS - 1)
#define GAMMAF  0.99f

#define TILE_FLOATS (RPB * ATOMS)       // 6528 floats = 26112 bytes
#define TILE_CHUNKS (TILE_FLOATS / 4)   // 1632 x 16B chunks

__global__ __launch_bounds__(RPB)
void c51_project_kernel(const float* __restrict__ reward,
                        const float* __restrict__ dist,
                        const float* __restrict__ supports,
                        const int*   __restrict__ mask,
                        float* __restrict__ out)
{
    // single LDS block: [ dist tile | accumulator tile | supports ]
    __shared__ __align__(16) float smem[TILE_FLOATS * 2 + 64];
    float* sdist = smem;
    float* sacc  = smem + TILE_FLOATS;
    float* ssup  = smem + 2 * TILE_FLOATS;

    const uint32_t tid = threadIdx.x;
    const uint32_t r0  = blockIdx.x * RPB;
    const uint32_t row = r0 + tid;

    const uint32_t sdist_base = (uint32_t)(uintptr_t)sdist;   // LDS byte offset
    const uint32_t sacc_base  = (uint32_t)(uintptr_t)sacc;
    const uint32_t ssup_base  = (uint32_t)(uintptr_t)ssup;

    // ---- stage dist tile into LDS: async, coalesced 16B chunks ----
    const float* gdist = dist + (size_t)r0 * ATOMS;           // 16B aligned
    #pragma unroll
    for (uint32_t k = 0; k < 13; ++k) {
        uint32_t c = tid + k * RPB;
        if (c < TILE_CHUNKS) {
            uint32_t boff  = c * 16u;
            uint32_t laddr = sdist_base + boff;
            asm volatile("global_load_async_to_lds_b128 %0, %1, %2"
                         :: "v"(laddr), "v"(boff), "s"(gdist) : "memory");
        }
    }
    if (tid < ATOMS) {
        uint32_t boff  = tid * 4u;
        uint32_t laddr = ssup_base + boff;
        asm volatile("global_load_async_to_lds_b32 %0, %1, %2"
                     :: "v"(boff == 0xFFFFFFFFu ? 0u : laddr), "v"(boff),
                        "s"(supports) : "memory");
    }

    // per-row scalars (regular loads, overlap with async transfers)
    const float r  = reward[row];
    const float cs = GAMMAF * (float)mask[row];   // gamma^n * mask

    // zero own accumulator row (thread-private; same-wave DS ops are in-order,
    // so no barrier needed between this and our own scatter below)
    float* myacc = &sacc[tid * ATOMS];
    #pragma unroll
    for (int a = 0; a < ATOMS; ++a) myacc[a] = 0.0f;

    asm volatile("s_wait_asynccnt 0" ::: "memory");
    __syncthreads();

    // ---- projection + row-local scatter ----
    const float* myrow = &sdist[tid * ATOMS];     // word stride 51: conflict-free
    for (int a = 0; a < ATOMS; ++a) {
        float d  = myrow[a];
        float tz = r + cs * ssup[a];
        tz       = fminf(fmaxf(tz, VMINF), VMAXF);
        float b  = (tz - VMINF) / DELTAF;
        int   l  = (int)floorf(b);
        int   u  = (int)ceilf(b);
        if (u > 0 && l == u)            l -= 1;   // sequential disambiguation,
        if (l < ATOMS - 1 && l == u)    u += 1;   // matches reference order
        float wl = d * ((float)u - b);
        float wu = d * (b - (float)l);
        atomicAdd(&myacc[l], wl);                 // ds_add_f32 (no-return)
        atomicAdd(&myacc[u], wu);
    }

    __syncthreads();

    // ---- async store accumulator tile -> out (contiguous, coalesced b128) ----
    float* gout = out + (size_t)r0 * ATOMS;
    #pragma unroll
    for (uint32_t k = 0; k < 13; ++k) {
        uint32_t c = tid + k * RPB;
        if (c < TILE_CHUNKS) {
            uint32_t boff  = c * 16u;
            uint32_t laddr = sacc_base + boff;
            asm volatile("global_store_async_from_lds_b128 %0, %1, %2"
                         :: "v"(boff), "v"(laddr), "s"(gout) : "memory");
        }
    }
    asm volatile("s_wait_asynccnt 0" ::: "memory");
}

extern "C" void kernel_launch(void* const* d_in, const int* in_sizes, int n_in,
                              void* d_out, int out_size, void* d_ws, size_t ws_size,
                              hipStream_t stream) {
    const float* reward   = (const float*)d_in[0];   // [B]
    const float* dist     = (const float*)d_in[1];   // [B, 51]
    const float* supports = (const float*)d_in[2];   // [51]
    const int*   mask     = (const int*)d_in[3];     // [B]
    float* out = (float*)d_out;                      // [B, 51]

    const int B = in_sizes[0];                       // 1048576 (multiple of RPB)
    const int blocks = B / RPB;

    hipLaunchKernelGGL(c51_project_kernel, dim3(blocks), dim3(RPB), 0, stream,
                       reward, dist, supports, mask, out);
}